// SimplifiedAfterShockGNN_44212393345479
// MI455X (gfx1250) — compile-verified
//
#include <hip/hip_runtime.h>
#include <cstddef>

#define NNODE 50000
#define NEDGE 800000
#define MDIM  16
#define WDIM  512
#define HDIM  128
#define NHEAD 2
#define NLAYER 3
#define NGRAPH 64
#define BN_EPS 1e-5f
#define NEG_SLOPE 0.2f
#define KEY_NEGINF 0x007FFFFFu   // ordered-key encoding of -inf

typedef __attribute__((ext_vector_type(2))) float v2f;
typedef __attribute__((ext_vector_type(8))) float v8f;

// ---- ordered-key float <-> uint (monotone encoding for atomicMax) ----
__device__ __forceinline__ unsigned fkey(float f) {
    unsigned b = __float_as_uint(f);
    return (b & 0x80000000u) ? ~b : (b | 0x80000000u);
}
__device__ __forceinline__ float keyf(unsigned k) {
    unsigned b = (k & 0x80000000u) ? (k ^ 0x80000000u) : ~k;
    return __uint_as_float(b);
}

// =====================================================================
// fp32 GEMM via V_WMMA_F32_16X16X4_F32 with 4-tile N-blocking.
// One wave computes a 16x64 strip of C = A(MxK) * B(KxNC) [+bias][relu]:
// the A fragment (ISA 16x4 layout: lanes 0-15 rows, v0/v1 = K,K+1 lower
// half-wave / K+2,K+3 upper half-wave) is reused across 4 WMMAs, cutting
// A re-reads from L2 by 4x and raising wmma-per-load density.
// Requires M%16==0 and NC%64==0 (true at every call site).
// =====================================================================
__global__ void gemm16_f32_wmma(const float* __restrict__ A, int lda,
                                const float* __restrict__ B, int ldb,
                                const float* __restrict__ bias,
                                float* __restrict__ C, int ldc, int col_off,
                                int M, int K, int NC, int do_relu)
{
    const int wave   = blockIdx.x * (blockDim.x >> 5) + (threadIdx.x >> 5);
    const int lane   = threadIdx.x & 31;
    const int nstrip = NC >> 6;                 // 64-column strips
    const int strips = (M >> 4) * nstrip;
    if (wave >= strips) return;                 // wave-uniform: EXEC stays all-1s
    const int tm   = wave / nstrip;
    const int tn   = wave - tm * nstrip;
    const int half = lane >> 4;                 // 0: K,K+1   1: K+2,K+3
    const int l    = lane & 15;

    const float* arow  = A + (size_t)(tm * 16 + l) * lda;
    const float* bbase = B + (tn * 64 + l);

    v8f acc0 = {}, acc1 = {}, acc2 = {}, acc3 = {};
    for (int k = 0; k < K; k += 4) {
        const int ka = k + half * 2;
        v2f a;
        a.x = arow[ka];
        a.y = arow[ka + 1];
        const float* b0 = bbase + (size_t)ka * ldb;
        const float* b1 = bbase + (size_t)(ka + 1) * ldb;
        v2f b;
        b.x = b0[0];  b.y = b1[0];
        acc0 = __builtin_amdgcn_wmma_f32_16x16x4_f32(false, a, false, b, (short)0, acc0, false, false);
        b.x = b0[16]; b.y = b1[16];
        acc1 = __builtin_amdgcn_wmma_f32_16x16x4_f32(false, a, false, b, (short)0, acc1, false, false);
        b.x = b0[32]; b.y = b1[32];
        acc2 = __builtin_amdgcn_wmma_f32_16x16x4_f32(false, a, false, b, (short)0, acc2, false, false);
        b.x = b0[48]; b.y = b1[48];
        acc3 = __builtin_amdgcn_wmma_f32_16x16x4_f32(false, a, false, b, (short)0, acc3, false, false);
    }

    // C/D layout: VGPR r -> row (tm*16 + half*8 + r), lanes' l -> column
    float* crow0 = C + (size_t)(tm * 16 + half * 8) * ldc + col_off + tn * 64 + l;
    v8f accs[4] = {acc0, acc1, acc2, acc3};
#pragma unroll
    for (int j = 0; j < 4; ++j) {
        const float bv = bias ? bias[tn * 64 + j * 16 + l] : 0.0f;
        float* crow = crow0 + j * 16;
#pragma unroll
        for (int r = 0; r < 8; ++r) {
            float v = accs[j][r] + bv;
            if (do_relu) v = fmaxf(v, 0.0f);
            crow[(size_t)r * ldc] = v;
        }
    }
}

// ---------------- fills ----------------
__global__ void fill_f32(float* __restrict__ p, float v, size_t n) {
    for (size_t i = blockIdx.x * (size_t)blockDim.x + threadIdx.x; i < n;
         i += (size_t)gridDim.x * blockDim.x) p[i] = v;
}
__global__ void fill_u32(unsigned* __restrict__ p, unsigned v, size_t n) {
    for (size_t i = blockIdx.x * (size_t)blockDim.x + threadIdx.x; i < n;
         i += (size_t)gridDim.x * blockDim.x) p[i] = v;
}

// ---------------- BatchNorm (training-mode, over node dim) ----------------
__global__ void bn_stats(const float* __restrict__ x, int C, int n,
                         int rows_per_block, float* __restrict__ gsum,
                         float* __restrict__ gsq)
{
    __shared__ float ssum[256];
    __shared__ float ssq[256];
    const int t = threadIdx.x;
    if (t < C) { ssum[t] = 0.f; ssq[t] = 0.f; }
    __syncthreads();
    const int col  = t % C;
    const int roff = t / C;
    const int rstr = blockDim.x / C;
    const int r0 = blockIdx.x * rows_per_block;
    int r1 = r0 + rows_per_block; if (r1 > n) r1 = n;
    float ls = 0.f, lq = 0.f;
    for (int r = r0 + roff; r < r1; r += rstr) {
        float v = x[(size_t)r * C + col];
        ls += v; lq += v * v;
    }
    atomicAdd(&ssum[col], ls);
    atomicAdd(&ssq[col], lq);
    __syncthreads();
    if (t < C) {
        atomicAdd(&gsum[t], ssum[t]);
        atomicAdd(&gsq[t], ssq[t]);
    }
}

__global__ void bn_finalize(const float* __restrict__ gsum,
                            const float* __restrict__ gsq, int C, float invn,
                            float* __restrict__ mean, float* __restrict__ istd)
{
    int c = blockIdx.x * blockDim.x + threadIdx.x;
    if (c < C) {
        float m = gsum[c] * invn;
        float v = gsq[c] * invn - m * m;
        mean[c] = m;
        istd[c] = rsqrtf(v + BN_EPS);
    }
}

// y = relu(gamma*(x-mean)*istd + beta [+ res])   (in-place safe)
__global__ void bn_apply(const float* __restrict__ x, const float* __restrict__ res,
                         const float* __restrict__ mean, const float* __restrict__ istd,
                         const float* __restrict__ gamma, const float* __restrict__ beta,
                         float* __restrict__ out, int C, size_t total)
{
    for (size_t idx = blockIdx.x * (size_t)blockDim.x + threadIdx.x; idx < total;
         idx += (size_t)gridDim.x * blockDim.x) {
        int c = (int)(idx % C);
        float v = (x[idx] - mean[c]) * istd[c] * gamma[c] + beta[c];
        if (res) v += res[idx];
        out[idx] = fmaxf(v, 0.0f);
    }
}

// ---------------- GAT attention scores: per (node, head) dot products -----
__global__ void attn_scores(const float* __restrict__ h,
                            const float* __restrict__ att_src,
                            const float* __restrict__ att_dst,
                            float* __restrict__ asrc, float* __restrict__ adst)
{
    const int w = blockIdx.x * (blockDim.x >> 5) + (threadIdx.x >> 5);
    const int lane = threadIdx.x & 31;
    if (w >= NNODE * NHEAD) return;
    const int node = w >> 1;
    const int head = w & 1;
    const float* hp = h + (size_t)node * (NHEAD * HDIM) + head * HDIM;
    const float* as = att_src + head * HDIM;
    const float* ad = att_dst + head * HDIM;
    float s0 = 0.f, s1 = 0.f;
    for (int c = lane; c < HDIM; c += 32) {
        float hv = hp[c];
        s0 += hv * as[c];
        s1 += hv * ad[c];
    }
    for (int off = 16; off > 0; off >>= 1) {
        s0 += __shfl_xor(s0, off, 32);
        s1 += __shfl_xor(s1, off, 32);
    }
    if (lane == 0) { asrc[w] = s0; adst[w] = s1; }
}

// ---------------- edge passes (segment softmax + aggregation) ----------------
__global__ void edge_max(const float* __restrict__ asrc, const float* __restrict__ adst,
                         const int* __restrict__ src, const int* __restrict__ dst,
                         float* __restrict__ ebuf, unsigned* __restrict__ mkey)
{
    int e = blockIdx.x * blockDim.x + threadIdx.x;
    if (e >= NEDGE) return;
    int s = src[e], d = dst[e];
#pragma unroll
    for (int h = 0; h < NHEAD; ++h) {
        float v = asrc[s * NHEAD + h] + adst[d * NHEAD + h];
        v = (v > 0.f) ? v : NEG_SLOPE * v;       // leaky relu
        ebuf[(size_t)e * NHEAD + h] = v;
        atomicMax(&mkey[d * NHEAD + h], fkey(v));
    }
}

__global__ void edge_expsum(float* __restrict__ ebuf, const unsigned* __restrict__ mkey,
                            const int* __restrict__ dst, float* __restrict__ sbuf)
{
    int e = blockIdx.x * blockDim.x + threadIdx.x;
    if (e >= NEDGE) return;
    int d = dst[e];
#pragma unroll
    for (int h = 0; h < NHEAD; ++h) {
        float m  = keyf(mkey[d * NHEAD + h]);
        float ex = __expf(ebuf[(size_t)e * NHEAD + h] - m);
        ebuf[(size_t)e * NHEAD + h] = ex;
        atomicAdd(&sbuf[d * NHEAD + h], ex);
    }
}

// one block (256 threads) per edge: all HEADS*H channels
__global__ void edge_scatter(const float* __restrict__ ebuf, const float* __restrict__ sbuf,
                             const int* __restrict__ src, const int* __restrict__ dst,
                             const float* __restrict__ h, float* __restrict__ agg)
{
    const int e = blockIdx.x;
    const int c = threadIdx.x;                   // 0..255
    const int s = src[e], d = dst[e];
    const int hd = c >> 7;
    float alpha = ebuf[(size_t)e * NHEAD + hd] / (sbuf[d * NHEAD + hd] + 1e-16f);
    float v = h[(size_t)s * (NHEAD * HDIM) + c] * alpha;
    atomicAdd(&agg[(size_t)d * (NHEAD * HDIM) + c], v);
}

// ---------------- graph mean pooling ----------------
__global__ void pool_accum(const float* __restrict__ x, const int* __restrict__ batch,
                           float* __restrict__ xg, float* __restrict__ counts)
{
    const size_t total = (size_t)NNODE * (NHEAD * HDIM);
    for (size_t idx = blockIdx.x * (size_t)blockDim.x + threadIdx.x; idx < total;
         idx += (size_t)gridDim.x * blockDim.x) {
        int n = (int)(idx >> 8);
        int c = (int)(idx & 255);
        int g = batch[n];
        atomicAdd(&xg[g * (NHEAD * HDIM) + c], x[idx]);
        if (c == 0) atomicAdd(&counts[g], 1.0f);
    }
}

__global__ void pool_div(float* __restrict__ xg, const float* __restrict__ counts)
{
    int idx = blockIdx.x * blockDim.x + threadIdx.x;
    if (idx < NGRAPH * NHEAD * HDIM)
        xg[idx] /= fmaxf(counts[idx >> 8], 1.0f);
}

// ---------------- output heads: [64,128] @ [128,1] + b ----------------
__global__ void head_out(const float* __restrict__ hid, const float* __restrict__ W2,
                         const float* __restrict__ b2, float* __restrict__ out)
{
    __shared__ float red[128];
    const int g = blockIdx.x;
    const int t = threadIdx.x;
    red[t] = hid[(size_t)g * HDIM + t] * W2[t];
    __syncthreads();
    for (int s = 64; s > 0; s >>= 1) {
        if (t < s) red[t] += red[t + s];
        __syncthreads();
    }
    if (t == 0) out[g] = red[0] + b2[0];
}

// =====================================================================
extern "C" void kernel_launch(void* const* d_in, const int* in_sizes, int n_in,
                              void* d_out, int out_size, void* d_ws, size_t ws_size,
                              hipStream_t stream)
{
    // ---- inputs (setup_inputs dict order; gat_Ws list expands to 3 entries)
    const float* metadata = (const float*)d_in[0];
    const float* wavef    = (const float*)d_in[1];
    const int*   eidx     = (const int*)  d_in[2];
    const int*   batch    = (const int*)  d_in[3];
    const float* meta_W   = (const float*)d_in[4];
    const float* meta_b   = (const float*)d_in[5];
    const float* wave_W   = (const float*)d_in[6];
    const float* wave_b   = (const float*)d_in[7];
    const float* comb_W   = (const float*)d_in[8];
    // comb_b (d_in[9]) cancels inside training-mode BN (constant mean shift)
    const float* comb_gamma = (const float*)d_in[10];
    const float* comb_beta  = (const float*)d_in[11];
    const float* gat_W[NLAYER] = { (const float*)d_in[12], (const float*)d_in[13],
                                   (const float*)d_in[14] };
    const float* att_src = (const float*)d_in[15];   // [L, HEADS, H]
    const float* att_dst = (const float*)d_in[16];
    // gat_bias (d_in[17]) cancels inside BN as well
    const float* bn_gamma = (const float*)d_in[18];  // [L, 256]
    const float* bn_beta  = (const float*)d_in[19];
    const float* lat_W1 = (const float*)d_in[20];
    const float* lat_b1 = (const float*)d_in[21];
    const float* lat_W2 = (const float*)d_in[22];
    const float* lat_b2 = (const float*)d_in[23];
    const float* lon_W1 = (const float*)d_in[24];
    const float* lon_b1 = (const float*)d_in[25];
    const float* lon_W2 = (const float*)d_in[26];
    const float* lon_b2 = (const float*)d_in[27];
    (void)in_sizes; (void)n_in; (void)out_size; (void)ws_size;

    const int* src = eidx;
    const int* dst = eidx + NEDGE;
    float* out = (float*)d_out;

    // ---- workspace carve (floats)
    float* W = (float*)d_ws;
    size_t off = 0;
    auto alloc = [&](size_t n) { float* p = W + off; off += n; return p; };
    const size_t NB = (size_t)NNODE * 256;
    float* B0   = alloc(NB);            // encoder concat -> rotates as agg/x
    float* B1   = alloc(NB);            // x after comb   -> rotates as x/agg
    float* BH   = alloc(NB);            // h = x @ W
    float* ebuf = alloc((size_t)NEDGE * NHEAD);
    unsigned* mkey = (unsigned*)alloc((size_t)NNODE * NHEAD);
    float* sbuf = alloc((size_t)NNODE * NHEAD);
    float* asrc = alloc((size_t)NNODE * NHEAD);
    float* adst = alloc((size_t)NNODE * NHEAD);
    float* gsum = alloc(256);
    float* gsq  = alloc(256);
    float* mean = alloc(256);
    float* istd = alloc(256);
    float* xg   = alloc((size_t)NGRAPH * 256);
    float* cnts = alloc(NGRAPH);
    float* hid  = alloc((size_t)NGRAPH * HDIM);

    auto gemm = [&](const float* A, int lda, const float* Bm, int ldb,
                    const float* bias, float* C, int ldc, int coloff,
                    int M, int K, int NC, int relu) {
        int strips = (M / 16) * (NC / 64);   // one 16x64 strip per wave
        int blocks = (strips + 7) / 8;       // 8 waves per 256-thread block
        gemm16_f32_wmma<<<blocks, 256, 0, stream>>>(A, lda, Bm, ldb, bias, C, ldc,
                                                    coloff, M, K, NC, relu);
    };
    auto zeroF = [&](float* p, size_t n) {
        fill_f32<<<2048, 256, 0, stream>>>(p, 0.0f, n);
    };

    const int EB = (NEDGE + 255) / 256;
    const int BN_BLK = 256;
    const int RPB = (NNODE + BN_BLK - 1) / BN_BLK;

    // ---- encoders: xcat = [relu(meta@W+b) | relu(wave@W+b)]  -> B0 [N,256]
    gemm(metadata, MDIM, meta_W, HDIM, meta_b, B0, 256, 0,    NNODE, MDIM, HDIM, 1);
    gemm(wavef,    WDIM, wave_W, HDIM, wave_b, B0, 256, HDIM, NNODE, WDIM, HDIM, 1);

    // ---- combiner: B1 = relu(bn(xcat @ comb_W))  [N,128]
    gemm(B0, 256, comb_W, HDIM, nullptr, B1, HDIM, 0, NNODE, 256, HDIM, 0);
    zeroF(gsum, 256); zeroF(gsq, 256);
    bn_stats<<<BN_BLK, 256, 0, stream>>>(B1, HDIM, NNODE, RPB, gsum, gsq);
    bn_finalize<<<1, 256, 0, stream>>>(gsum, gsq, HDIM, 1.0f / NNODE, mean, istd);
    bn_apply<<<2048, 256, 0, stream>>>(B1, nullptr, mean, istd, comb_gamma, comb_beta,
                                       B1, HDIM, (size_t)NNODE * HDIM);

    // ---- GAT layers (x/agg buffers rotate; BN applied in-place on agg)
    float* x   = B1;
    float* agg = B0;
    int cin = HDIM;
    for (int i = 0; i < NLAYER; ++i) {
        // h = x @ W  [N, 256]
        gemm(x, cin, gat_W[i], NHEAD * HDIM, nullptr, BH, NHEAD * HDIM, 0,
             NNODE, cin, NHEAD * HDIM, 0);
        // attention scores
        attn_scores<<<(NNODE * NHEAD + 3) / 4, 128, 0, stream>>>(
            BH, att_src + i * NHEAD * HDIM, att_dst + i * NHEAD * HDIM, asrc, adst);
        // segment softmax over incoming edges
        fill_u32<<<64, 256, 0, stream>>>(mkey, KEY_NEGINF, (size_t)NNODE * NHEAD);
        zeroF(sbuf, (size_t)NNODE * NHEAD);
        zeroF(agg, NB);
        edge_max<<<EB, 256, 0, stream>>>(asrc, adst, src, dst, ebuf, mkey);
        edge_expsum<<<EB, 256, 0, stream>>>(ebuf, mkey, dst, sbuf);
        edge_scatter<<<NEDGE, 256, 0, stream>>>(ebuf, sbuf, src, dst, BH, agg);
        // BN (+residual for i>0) + relu, in place on agg
        zeroF(gsum, 256); zeroF(gsq, 256);
        bn_stats<<<BN_BLK, 256, 0, stream>>>(agg, NHEAD * HDIM, NNODE, RPB, gsum, gsq);
        bn_finalize<<<1, 256, 0, stream>>>(gsum, gsq, NHEAD * HDIM, 1.0f / NNODE, mean, istd);
        bn_apply<<<2048, 256, 0, stream>>>(agg, (i > 0) ? x : nullptr, mean, istd,
                                           bn_gamma + i * NHEAD * HDIM,
                                           bn_beta + i * NHEAD * HDIM,
                                           agg, NHEAD * HDIM, NB);
        float* t = x; x = agg; agg = t;
        cin = NHEAD * HDIM;
    }

    // ---- mean pooling per graph
    zeroF(xg, (size_t)NGRAPH * 256); zeroF(cnts, NGRAPH);
    pool_accum<<<2048, 256, 0, stream>>>(x, batch, xg, cnts);
    pool_div<<<(NGRAPH * 256 + 255) / 256, 256, 0, stream>>>(xg, cnts);

    // ---- output heads (WMMA for the 64x256x128 layer, reduction for the last)
    gemm(xg, 256, lat_W1, HDIM, lat_b1, hid, HDIM, 0, NGRAPH, 256, HDIM, 1);
    head_out<<<NGRAPH, 128, 0, stream>>>(hid, lat_W2, lat_b2, out);
    gemm(xg, 256, lon_W1, HDIM, lon_b1, hid, HDIM, 0, NGRAPH, 256, HDIM, 1);
    head_out<<<NGRAPH, 128, 0, stream>>>(hid, lon_W2, lon_b2, out + NGRAPH);
}